// CenterLoss_76897094467952
// MI455X (gfx1250) — compile-verified
//
#include <hip/hip_runtime.h>

typedef __attribute__((ext_vector_type(2))) float v2f;
typedef __attribute__((ext_vector_type(8))) float v8f;

#define NCLASS 1024

__device__ __forceinline__ void atomic_add_f32(float* p, float v) {
    __hip_atomic_fetch_add(p, v, __ATOMIC_RELAXED, __HIP_MEMORY_SCOPE_AGENT);
}
__device__ __forceinline__ void atomic_add_f64(double* p, double v) {
    __hip_atomic_fetch_add(p, v, __ATOMIC_RELAXED, __HIP_MEMORY_SCOPE_AGENT);
}

// ---------------- zero workspace ----------------
__global__ void cl_zero(unsigned* ws, int nwords) {
    int i = blockIdx.x * blockDim.x + threadIdx.x;
    if (i < nwords) ws[i] = 0u;
}

// ---------------- pass 1: stream X, scatter class sums, sum of squares ----
// One wave (32 lanes) owns a sample row at a time. Lanes read consecutive
// float4s (512 B per wave-iteration, fully coalesced), scatter 4 f32 atomics
// each into the class row (2 MB target, L2-resident), and privately
// accumulate ||x||^2, reduced per-block into one f64 atomic.
__global__ void cl_pass1(const float* __restrict__ x,
                         const int* __restrict__ tgt,
                         float* __restrict__ sums,
                         unsigned* __restrict__ counts,
                         double* __restrict__ acc,
                         int nsamples, int dims) {
    const int lane   = threadIdx.x & 31;
    const int wave   = (blockIdx.x * blockDim.x + threadIdx.x) >> 5;
    const int nwaves = (gridDim.x * blockDim.x) >> 5;
    const int nvec   = dims >> 2;            // float4s per row

    double sq = 0.0;
    for (int s = wave; s < nsamples; s += nwaves) {
        const int c = tgt[s];
        const float4* row  = (const float4*)(x + (size_t)s * dims);
        float*        srow = sums + (size_t)c * dims;
        if (lane == 0) atomicAdd(&counts[c], 1u);
        for (int d4 = lane; d4 < nvec; d4 += 32) {
            float4 v = row[d4];
            sq += (double)(v.x * v.x + v.y * v.y + v.z * v.z + v.w * v.w);
            const int d = d4 << 2;
            atomic_add_f32(&srow[d + 0], v.x);
            atomic_add_f32(&srow[d + 1], v.y);
            atomic_add_f32(&srow[d + 2], v.z);
            atomic_add_f32(&srow[d + 3], v.w);
        }
    }

    __shared__ double red[256];
    red[threadIdx.x] = sq;
    __syncthreads();
    for (int off = blockDim.x >> 1; off > 0; off >>= 1) {
        if ((int)threadIdx.x < off) red[threadIdx.x] += red[threadIdx.x + off];
        __syncthreads();
    }
    if (threadIdx.x == 0) atomic_add_f64(&acc[0], red[0]);
}

// ---------------- pass 2: WMMA Gram diagonal -> sum_c ||s_c||^2 / n_c -----
// One wave per 16-class tile. v_wmma_f32_16x16x4_f32 with
//   A[m,k] = S[tile*16+m, d0+k],  B[k,n] = S[tile*16+n, d0+k]
// so D[m,n] += sum_k S_m S_n; the diagonal accumulates ||s_c||^2.
// Per the ISA f32 operand layouts, A and B have identical per-lane register
// images: lanes 0-15 hold (K=0,K=1), lanes 16-31 hold (K=2,K=3) of row
// M/N = lane%16, so a single float2 load feeds both operands.
__global__ void cl_pass2(const float* __restrict__ sums,
                         const unsigned* __restrict__ counts,
                         double* __restrict__ acc,
                         int dims) {
    const int tile = blockIdx.x;             // 16-class tile
    const int lane = threadIdx.x;            // 0..31, single wave
    const int cls  = tile * 16 + (lane & 15);
    const int dofs = (lane >> 4) << 1;       // 0 for lanes 0-15, 2 for 16-31
    const float* base = sums + (size_t)cls * dims + dofs;

    v8f c = {};
    for (int d0 = 0; d0 < dims; d0 += 4) {
        v2f ab = *(const v2f*)(base + d0);
        c = __builtin_amdgcn_wmma_f32_16x16x4_f32(
                /*neg_a=*/false, ab, /*neg_b=*/false, ab,
                /*c_mod=*/(short)0, c, /*reuse_a=*/false, /*reuse_b=*/false);
    }

    // C/D layout: VGPR r, lanes 0-15 -> (M=r, N=lane); lanes 16-31 -> (M=r+8,
    // N=lane-16). Diagonal M==N lives in: lane r, VGPR r (M=N=r) and
    // lane 24+r, VGPR r (M=N=r+8).
    union { v8f v; float f[8]; } u;
    u.v = c;
    int myclass = -1;
    float diag = 0.0f;
    if (lane < 8)        { diag = u.f[lane];      myclass = tile * 16 + lane; }
    else if (lane >= 24) { diag = u.f[lane - 24]; myclass = tile * 16 + (lane - 24) + 8; }
    if (myclass >= 0) {
        const unsigned n = counts[myclass];
        const double dn = (n > 0u) ? (double)n : 1.0;
        atomic_add_f64(&acc[1], (double)diag / dn);
    }
}

// ---------------- finalize: loss = 0.5 * (sum||x||^2 - sum_c ||s_c||^2/n_c)
__global__ void cl_finalize(const double* __restrict__ acc, float* __restrict__ out) {
    if (threadIdx.x == 0 && blockIdx.x == 0)
        out[0] = (float)(0.5 * (acc[0] - acc[1]));
}

extern "C" void kernel_launch(void* const* d_in, const int* in_sizes, int n_in,
                              void* d_out, int out_size, void* d_ws, size_t ws_size,
                              hipStream_t stream) {
    const float* x   = (const float*)d_in[0];
    const int*   tgt = (const int*)d_in[1];
    // d_in[2] = num_classes (device scalar); reference fixes C = 1024.
    float* out = (float*)d_out;

    const int nsamples = in_sizes[1];                 // 65536
    const int dims     = in_sizes[0] / nsamples;      // 512

    // Workspace layout: sums [C*D f32] | counts [C u32] | acc [2 f64]
    const size_t sums_bytes   = (size_t)NCLASS * dims * sizeof(float);
    const size_t counts_bytes = (size_t)NCLASS * sizeof(unsigned);
    float*    sums   = (float*)d_ws;
    unsigned* counts = (unsigned*)((char*)d_ws + sums_bytes);
    double*   acc    = (double*)((char*)d_ws + sums_bytes + counts_bytes);

    const int nwords = (int)((sums_bytes + counts_bytes + 2 * sizeof(double)) / 4);
    cl_zero<<<(nwords + 255) / 256, 256, 0, stream>>>((unsigned*)d_ws, nwords);

    cl_pass1<<<2048, 256, 0, stream>>>(x, tgt, sums, counts, acc, nsamples, dims);

    cl_pass2<<<NCLASS / 16, 32, 0, stream>>>(sums, counts, acc, dims);

    cl_finalize<<<1, 32, 0, stream>>>(acc, out);
    (void)n_in; (void)out_size; (void)ws_size;
}